// Fuzzy_Router_72593537237142
// MI455X (gfx1250) — compile-verified
//
#include <hip/hip_runtime.h>
#include <cstdint>

// ---------------------------------------------------------------------------
// Shapes (fixed by setup_inputs): x (B=8, S=4096, D=4096) fp32,
// question_mask (B, Q=2048) — only its SHAPE matters (Q). T = Q-1 = 2047.
// Output: (B, S, 2) fp32.
// ---------------------------------------------------------------------------
#define DIM   4096
#define SEQ   4096
#define QLEN  2048
#define CHUNK 16           // rows per block in fused kernel
#define NCHK  (QLEN / CHUNK)   // 128 chunks per batch

// ---- fuzzy rule tables (compile-time) -------------------------------------
__constant__ float c_centers[15][3] = {
    {0.165f, 0.165f, 0.165f}, {0.165f, 0.165f, 0.835f}, {0.165f, 0.500f, 0.500f},
    {0.165f, 0.835f, 0.165f}, {0.165f, 0.835f, 0.835f}, {0.500f, 0.165f, 0.165f},
    {0.500f, 0.165f, 0.835f}, {0.500f, 0.500f, 0.500f}, {0.500f, 0.835f, 0.165f},
    {0.500f, 0.835f, 0.835f}, {0.835f, 0.165f, 0.165f}, {0.835f, 0.165f, 0.835f},
    {0.835f, 0.500f, 0.500f}, {0.835f, 0.835f, 0.165f}, {0.835f, 0.835f, 0.835f}};
__constant__ float c_conseq[15] = {
    0.0f, 0.333f, 0.333f, 0.333f, 0.667f, 0.333f, 0.667f, 0.5f,
    0.667f, 1.0f, 0.667f, 1.0f, 1.0f, 1.167f, 1.5f};
// cons_tab indexed by b1*9+b2*3+b3
__constant__ float c_cons_tab[27] = {
    0.0f,  0.f, 0.333f,  0.f, 0.333f, 0.f, 0.333f, 0.f, 0.667f,
    0.333f, 0.f, 0.667f, 0.f, 0.5f,   0.f, 0.667f, 0.f, 1.0f,
    0.667f, 0.f, 1.0f,   0.f, 1.0f,   0.f, 1.167f, 0.f, 1.5f};
#define VALID_MASK 0x0556AB55u   // bits {0,2,4,6,8,9,11,13,15,17,18,20,22,24,26}

// ---- wave32 block reductions (256 threads = 8 waves) ----------------------
template <bool IS_MAX>
__device__ __forceinline__ float block_reduce(float v, float* red) {
#pragma unroll
    for (int m = 16; m > 0; m >>= 1) {
        float o = __shfl_xor(v, m, 32);
        v = IS_MAX ? fmaxf(v, o) : (v + o);
    }
    __syncthreads();                       // protect red reuse
    if ((threadIdx.x & 31) == 0) red[threadIdx.x >> 5] = v;
    __syncthreads();
    float r = red[0];
#pragma unroll
    for (int i = 1; i < 8; i++) r = IS_MAX ? fmaxf(r, red[i]) : (r + red[i]);
    return r;                              // identical order in every thread
}

// ---------------------------------------------------------------------------
// K1: fused per-row entropy + per-chunk column partial sums.
// One block per (batch, 16-row chunk). Rows staged to LDS via CDNA5 async
// global->LDS copies (ASYNCcnt path), then 3 LDS passes (max, sumexp, ent).
// ---------------------------------------------------------------------------
__global__ __launch_bounds__(256) void fr_ent_colsum_kernel(
    const float* __restrict__ x, float* __restrict__ ent,
    float* __restrict__ colpart) {
    __shared__ float row[DIM];
    __shared__ float red[8];
    const int blk   = blockIdx.x;      // B*NCHK blocks
    const int b     = blk / NCHK;
    const int chunk = blk % NCHK;
    const int t     = threadIdx.x;

    float colacc[16];
#pragma unroll
    for (int i = 0; i < 16; i++) colacc[i] = 0.f;

    for (int rr = 0; rr < CHUNK; ++rr) {
        const int r = chunk * CHUNK + rr;
        const float* src = x + ((size_t)b * SEQ + r) * DIM;
        // async copy 16 KB row into LDS: 4 x (256 lanes x b128)
#pragma unroll
        for (int i = 0; i < 4; i++) {
            const int e = (i * 256 + t) * 4;
            unsigned ldsOff = (unsigned)(uintptr_t)(&row[e]);
            asm volatile("global_load_async_to_lds_b128 %0, %1, off"
                         :: "v"(ldsOff), "v"(src + e) : "memory");
        }
        asm volatile("s_wait_asynccnt 0x0" ::: "memory");
        __syncthreads();

        if (r != 0) {   // txt rows only feed the column mean
#pragma unroll
            for (int i = 0; i < 16; i++) colacc[i] += row[t + i * 256];
        }
        // pass 1: max
        float m = -__builtin_inff();
#pragma unroll
        for (int i = 0; i < 16; i++) m = fmaxf(m, row[t + i * 256]);
        m = block_reduce<true>(m, red);
        // pass 2: sum exp
        float s = 0.f;
#pragma unroll
        for (int i = 0; i < 16; i++) s += __expf(row[t + i * 256] - m);
        s = block_reduce<false>(s, red);
        const float inv_s = 1.0f / s;
        // pass 3: -sum p*log(p+1e-8), faithful to reference
        float acc = 0.f;
#pragma unroll
        for (int i = 0; i < 16; i++) {
            float p = __expf(row[t + i * 256] - m) * inv_s;
            acc += p * __logf(p + 1e-8f);
        }
        acc = block_reduce<false>(acc, red);
        if (t == 0) ent[b * QLEN + r] = -acc;
        __syncthreads();   // all reads of `row` done before next async overwrite
    }

    if (colpart != nullptr) {
        float* cp = colpart + (size_t)blk * DIM;
#pragma unroll
        for (int i = 0; i < 16; i++) cp[t + i * 256] = colacc[i];
    }
}

// K2a (fast path): fold 128 chunk partials per (batch,col) -> column mean.
__global__ __launch_bounds__(256) void fr_colreduce_kernel(
    const float* __restrict__ colpart, float* __restrict__ bsum) {
    const int idx = blockIdx.x * 256 + threadIdx.x;   // B*DIM threads
    const int b = idx >> 12, c = idx & (DIM - 1);
    const float* cp = colpart + (size_t)b * NCHK * DIM + c;
    float acc = 0.f;
#pragma unroll 8
    for (int k = 0; k < NCHK; k++) {
        if ((threadIdx.x & 31) == 0)
            __builtin_prefetch(cp + (size_t)(k + 16) * DIM, 0, 1);
        acc += cp[(size_t)k * DIM];
    }
    bsum[idx] = acc * (1.0f / 2047.0f);
}

// K2b (fallback if ws too small): re-read x for column means.
__global__ __launch_bounds__(256) void fr_colsum_kernel(
    const float* __restrict__ x, float* __restrict__ bsum) {
    const int idx = blockIdx.x * 256 + threadIdx.x;   // B*DIM threads
    const int b = idx >> 12, c = idx & (DIM - 1);
    const float* p = x + ((size_t)b * SEQ + 1) * DIM + c;
    float acc = 0.f;
    for (int r0 = 0; r0 < QLEN - 1; r0 += 8) {
        if ((threadIdx.x & 31) == 0)
            __builtin_prefetch(p + (size_t)(r0 + 64) * DIM, 0, 1);
#pragma unroll
        for (int u = 0; u < 8; ++u) {
            int r = r0 + u;
            if (r < QLEN - 1) acc += p[(size_t)r * DIM];
        }
    }
    bsum[idx] = acc * (1.0f / 2047.0f);
}

// K3: per-batch cosine(img_row, mean_txt).
__global__ __launch_bounds__(256) void fr_cos_kernel(
    const float* __restrict__ x, const float* __restrict__ bsum,
    float* __restrict__ x3) {
    __shared__ float red[8];
    const int b = blockIdx.x;
    const float* a  = x + (size_t)b * SEQ * DIM;   // row 0 = img
    const float* bm = bsum + (size_t)b * DIM;
    const int t = threadIdx.x;
    float dot = 0.f, na = 0.f, nb = 0.f;
    for (int i = t; i < DIM; i += 256) {
        float av = a[i], bv = bm[i];
        dot += av * bv; na += av * av; nb += bv * bv;
    }
    dot = block_reduce<false>(dot, red);
    na  = block_reduce<false>(na, red);
    nb  = block_reduce<false>(nb, red);
    if (t == 0) {
        float dna = fmaxf(sqrtf(na), 1e-8f);
        float dnb = fmaxf(sqrtf(nb), 1e-8f);
        x3[b] = dot / (dna * dnb);
    }
}

// K4: global entropy maxima (img rows vs txt rows), single block.
__global__ __launch_bounds__(256) void fr_maxent_kernel(
    const float* __restrict__ ent, float* __restrict__ mx, int B) {
    __shared__ float red[8];
    const int t = threadIdx.x;
    float mi = -__builtin_inff(), mt = -__builtin_inff();
    for (int i = t; i < B * QLEN; i += 256) {
        float e = ent[i];
        if ((i & (QLEN - 1)) == 0) mi = fmaxf(mi, e);
        else                       mt = fmaxf(mt, e);
    }
    mi = block_reduce<true>(mi, red);
    mt = block_reduce<true>(mt, red);
    if (t == 0) { mx[0] = mi; mx[1] = mt; }
}

// K5: fuzzy rule evaluation + output assembly (covers full (B,S,2)).
__global__ __launch_bounds__(256) void fr_fuzzy_kernel(
    const float* __restrict__ ent, const float* __restrict__ x3,
    const float* __restrict__ mx, float* __restrict__ out) {
    const int idx = blockIdx.x * 256 + threadIdx.x;   // B*SEQ threads
    const int b = idx >> 12, s = idx & (SEQ - 1);
    float f0 = 0.f, f1 = 0.f;
    if (s >= 1 && s < QLEN) {
        const float x1v = ent[b * QLEN] / fmaxf(mx[0], 1e-6f);
        const float x2v = ent[b * QLEN + s] / fmaxf(mx[1], 1e-6f);
        const float x3v = x3[b];
        const float feat[3] = {x1v, x2v, x3v};
        int bi[3]; bool inr = true;
#pragma unroll
        for (int k = 0; k < 3; k++) {
            float v = feat[k];
            bi[k] = v < 0.33f ? 0 : (v < 0.67f ? 1 : 2);
            inr = inr && (v >= 0.f) && (v < 1.f);
        }
        const int code = bi[0] * 9 + bi[1] * 3 + bi[2];
        const bool matched = inr && ((VALID_MASK >> code) & 1u);
        // top-2 nearest centers (ties keep lower index first, like lax.top_k)
        float d1 = __builtin_inff(), d2 = __builtin_inff();
        int i1 = 0, i2 = 0;
#pragma unroll
        for (int i = 0; i < 15; i++) {
            float dx = x1v - c_centers[i][0];
            float dy = x2v - c_centers[i][1];
            float dz = x3v - c_centers[i][2];
            float d = sqrtf(dx * dx + dy * dy + dz * dz);
            if (d < d1)      { d2 = d1; i2 = i1; d1 = d; i1 = i; }
            else if (d < d2) { d2 = d;  i2 = i; }
        }
        const float den = d1 + d2;
        const float lam = (den != 0.f) ? (d1 / den) : 0.5f;
        const float fi  = (1.f - lam) * c_conseq[i1] + lam * c_conseq[i2];
        const float f   = matched ? c_cons_tab[code] : fi;
        f0 = f; f1 = 1.f - f;
    }
    out[(size_t)idx * 2]     = f0;
    out[(size_t)idx * 2 + 1] = f1;
}

// ---------------------------------------------------------------------------
extern "C" void kernel_launch(void* const* d_in, const int* in_sizes, int n_in,
                              void* d_out, int out_size, void* d_ws, size_t ws_size,
                              hipStream_t stream) {
    const float* x = (const float*)d_in[0];
    // d_in[1] (question_mask) contributes only its shape; values unused.
    const int B = in_sizes[0] / (SEQ * DIM);
    float* ws   = (float*)d_ws;
    // ws layout (floats): ent[B*QLEN] | bsum[B*DIM] | x3[B] | mx[2] | colpart
    float* ent     = ws;
    float* bsum    = ent + (size_t)B * QLEN;
    float* x3      = bsum + (size_t)B * DIM;
    float* mx      = x3 + B;
    float* colpart = mx + 2;
    const size_t base_floats  = (size_t)B * QLEN + (size_t)B * DIM + B + 2;
    const size_t fused_floats = base_floats + (size_t)B * NCHK * DIM;
    const bool fused = ws_size >= fused_floats * sizeof(float);

    if (fused) {
        fr_ent_colsum_kernel<<<B * NCHK, 256, 0, stream>>>(x, ent, colpart);
        fr_colreduce_kernel<<<(B * DIM) / 256, 256, 0, stream>>>(colpart, bsum);
    } else {
        fr_ent_colsum_kernel<<<B * NCHK, 256, 0, stream>>>(x, ent, nullptr);
        fr_colsum_kernel<<<(B * DIM) / 256, 256, 0, stream>>>(x, bsum);
    }
    fr_cos_kernel<<<B, 256, 0, stream>>>(x, bsum, x3);
    fr_maxent_kernel<<<1, 256, 0, stream>>>(ent, mx, B);
    fr_fuzzy_kernel<<<(B * SEQ) / 256, 256, 0, stream>>>(ent, x3, mx, (float*)d_out);
}